// FeatureDecouplingLoss_72533407695367
// MI455X (gfx1250) — compile-verified
//
#include <hip/hip_runtime.h>
#include <hip/hip_bf16.h>

typedef __attribute__((ext_vector_type(2))) float v2f;
typedef __attribute__((ext_vector_type(8))) float v8f;

#define D 64
#define C 1000
#define CHUNK 128          // member rows staged in LDS per pass
#define MAXM 1024          // member-list capacity per class
#define TPB 512            // 16 waves of 32

// ---------------------------------------------------------------------------
// Kernel 1: L2-normalize rows.  One wave32 per row (2 floats per lane).
// ---------------------------------------------------------------------------
__global__ void fdl_normalize(const float* __restrict__ x,
                              float* __restrict__ xn, int n) {
    int gwave = (int)((blockIdx.x * blockDim.x + threadIdx.x) >> 5);
    int lane  = (int)(threadIdx.x & 31);
    if (gwave >= n) return;
    const float2 v = ((const float2*)(x + (size_t)gwave * D))[lane];
    float ss = v.x * v.x + v.y * v.y;
    #pragma unroll
    for (int off = 16; off > 0; off >>= 1)
        ss += __shfl_xor(ss, off, 32);
    float scale = 1.0f / fmaxf(sqrtf(ss), 1e-12f);
    float2 o; o.x = v.x * scale; o.y = v.y * scale;
    ((float2*)(xn + (size_t)gwave * D))[lane] = o;
}

// ---------------------------------------------------------------------------
// Kernel 2: one workgroup (16 waves) per class.
//  Phase A: ordered ballot-compaction scan of labels -> member list (exact,
//           deterministic count).
//  Phase B: gather member rows to LDS in chunks; each wave owns one 16x16
//           tile (tr,tc) of M_c = X^T X and accumulates with f32 WMMA
//           (V_WMMA_F32_16X16X4_F32, K=4 samples/step, zero-padded tail).
//           dsq (diagonal term) computed from the same LDS data.
//  Phase C: block reduction -> per-class loss term, atomics to accumulators.
// ---------------------------------------------------------------------------
__global__ __launch_bounds__(TPB)
void fdl_class_gram(const float* __restrict__ xn,
                    const int* __restrict__ labels, int n,
                    float* __restrict__ gsum, float* __restrict__ guniq) {
    const int cls  = (int)blockIdx.x;
    const int tid  = (int)threadIdx.x;
    const int wid  = tid >> 5;
    const int lane = tid & 31;

    __shared__ int   s_members[MAXM];
    __shared__ __align__(16) float s_x[CHUNK * D];   // 32 KB staged rows
    __shared__ int   s_wcnt[16];
    __shared__ int   s_listSize;
    __shared__ float s_fro[16];
    __shared__ float s_dsq[TPB];

    if (tid == 0) s_listSize = 0;
    __syncthreads();

    // ---- Phase A: ordered compaction of member indices -------------------
    for (int base = 0; base < n; base += TPB) {
        int i = base + tid;
        bool pred = (i < n) && (labels[i] == cls);
        unsigned int ball = (unsigned int)__ballot(pred);   // wave32 mask
        if (lane == 0) s_wcnt[wid] = __popc(ball);
        __syncthreads();
        int prefix = 0;
        for (int w = 0; w < wid; ++w) prefix += s_wcnt[w];
        int pos = s_listSize + prefix + __popc(ball & ((1u << lane) - 1u));
        if (pred && pos < MAXM) s_members[pos] = i;
        __syncthreads();
        if (tid == 0) {
            int tot = 0;
            for (int w = 0; w < 16; ++w) tot += s_wcnt[w];
            s_listSize += tot;
        }
        __syncthreads();
    }
    const int n_c = min(s_listSize, MAXM);

    // ---- Phase B: WMMA Gram accumulation ---------------------------------
    const int tr = wid >> 2;          // tile row   (dims 16*tr .. 16*tr+15)
    const int tc = wid & 3;           // tile col   (dims 16*tc .. 16*tc+15)
    const int m    = lane & 15;
    const int koff = (lane >> 4) << 1;   // 0 for lanes 0-15, 2 for 16-31

    v8f acc = {0.f, 0.f, 0.f, 0.f, 0.f, 0.f, 0.f, 0.f};
    float dsqAcc = 0.f;

    for (int cs = 0; cs < n_c; cs += CHUNK) {
        int rows  = min(CHUNK, n_c - cs);
        int rowsP = (rows + 3) & ~3;

        // gather member rows (float4 granularity)
        for (int slot = tid; slot < rows * 16; slot += TPB) {
            int j = slot >> 4, q = slot & 15;
            int src = s_members[cs + j];
            ((float4*)s_x)[j * 16 + q] =
                ((const float4*)(xn + (size_t)src * D))[q];
        }
        // zero-pad K tail (<= 3 rows)
        for (int idx = tid; idx < (rowsP - rows) * D; idx += TPB)
            s_x[rows * D + idx] = 0.f;
        __syncthreads();

        // diagonal term: dsq_i = (sum_d xn^2)^2
        if (tid < rows) {
            float ss = 0.f;
            #pragma unroll
            for (int d = 0; d < D; ++d) {
                float v = s_x[tid * D + d];
                ss += v * v;
            }
            dsqAcc += ss * ss;
        }

        // f32 WMMA: M[16tr+a, 16tc+b] += sum_k X[k,16tr+a] * X[k,16tc+b]
        for (int k0 = 0; k0 < rowsP; k0 += 4) {
            v2f a, b;
            a.x = s_x[(k0 + koff    ) * D + tr * 16 + m];
            a.y = s_x[(k0 + koff + 1) * D + tr * 16 + m];
            b.x = s_x[(k0 + koff    ) * D + tc * 16 + m];
            b.y = s_x[(k0 + koff + 1) * D + tc * 16 + m];
            acc = __builtin_amdgcn_wmma_f32_16x16x4_f32(
                false, a, false, b, (short)0, acc, false, false);
        }
        __syncthreads();   // before next gather overwrites s_x
    }

    // ---- Phase C: reduce ||M||_F^2 and dsq, emit per-class term ----------
    float part = 0.f;
    #pragma unroll
    for (int t = 0; t < 8; ++t) part += acc[t] * acc[t];
    #pragma unroll
    for (int off = 16; off > 0; off >>= 1)
        part += __shfl_xor(part, off, 32);
    if (lane == 0) s_fro[wid] = part;
    s_dsq[tid] = dsqAcc;
    __syncthreads();

    if (tid == 0) {
        float fro = 0.f;
        for (int w = 0; w < 16; ++w) fro += s_fro[w];
        float dsum = 0.f;
        for (int t = 0; t < TPB; ++t) dsum += s_dsq[t];
        float offd = fro - dsum;
        if (n_c > 1)
            atomicAdd(gsum, offd / ((float)n_c * (float)D));
        if (n_c > 0)
            atomicAdd(guniq, 1.0f);
    }
}

// ---------------------------------------------------------------------------
// Kernel 3: finalize scalar.
// ---------------------------------------------------------------------------
__global__ void fdl_finalize(const float* __restrict__ accums,
                             float* __restrict__ out) {
    float u = accums[1];
    out[0] = (u > 0.f) ? accums[0] / u : 0.f;
}

extern "C" void kernel_launch(void* const* d_in, const int* in_sizes, int n_in,
                              void* d_out, int out_size, void* d_ws, size_t ws_size,
                              hipStream_t stream) {
    const float* feats  = (const float*)d_in[0];
    const int*   labels = (const int*)d_in[1];
    float*       out    = (float*)d_out;
    const int N = in_sizes[1];                 // labels count = samples

    float* xn     = (float*)d_ws;                                   // N*64 f32
    float* accums = (float*)((char*)d_ws + (size_t)N * D * sizeof(float)); // [gsum, guniq]

    hipMemsetAsync(accums, 0, 2 * sizeof(float), stream);

    fdl_normalize<<<(N + 7) / 8, 256, 0, stream>>>(feats, xn, N);
    fdl_class_gram<<<C, TPB, 0, stream>>>(xn, labels, N, accums, accums + 1);
    fdl_finalize<<<1, 1, 0, stream>>>(accums, out);
}